// MultiModalAttention_13168369730129
// MI455X (gfx1250) — compile-verified
//
#include <hip/hip_runtime.h>
#include <math.h>

typedef float v2f __attribute__((ext_vector_type(2)));
typedef float v8f __attribute__((ext_vector_type(8)));

#define NB    4
#define T_DIM 2048
#define L_DIM 2048
#define DQIN  32
#define DFEAT 34
#define DQK   64
#define KPAD  68   // 64 proj + 1 ones col + 3 zero pad -> multiple of wmma K=4
#define DVDIM 64

// ---------------- prep: Q = [ref_data @ Wq, 1, 0,0,0] ----------------
__global__ void prep_q_kernel(const float* __restrict__ ref_data,
                              const float* __restrict__ Wq,
                              float* __restrict__ Qpad) {
  int idx = blockIdx.x * blockDim.x + threadIdx.x;
  const int TOT = NB * T_DIM * KPAD;
  if (idx >= TOT) return;
  int c   = idx % KPAD;
  int row = idx / KPAD;            // n*T + t
  float v;
  if (c < DQK) {
    const float* rd = ref_data + (size_t)row * DQIN;
    float acc = 0.f;
    #pragma unroll
    for (int d = 0; d < DQIN; ++d) acc += rd[d] * Wq[d * DQK + c];
    v = acc;
  } else if (c == DQK) {
    v = 1.0f;
  } else {
    v = 0.0f;
  }
  Qpad[(size_t)row * KPAD + c] = v;
}

// -------- prep: K = [1, -(x*Wk+bk), 0,0,0],  V = data[:, :32] @ Wv + bv --------
__global__ void prep_kv_kernel(const float* __restrict__ data,
                               const float* __restrict__ Wk,
                               const float* __restrict__ bk,
                               const float* __restrict__ Wv,
                               const float* __restrict__ bv,
                               float* __restrict__ Kp,
                               float* __restrict__ Vm) {
  int idx = blockIdx.x * blockDim.x + threadIdx.x;
  const int CW  = KPAD + DVDIM;
  const int TOT = NB * L_DIM * CW;
  if (idx >= TOT) return;
  int c   = idx % CW;
  int row = idx / CW;              // n*L + l
  const float* drow = data + (size_t)row * DFEAT;
  if (c < KPAD) {
    float v;
    if (c == 0)          v = 1.0f;
    else if (c <= DQK)   v = -(drow[DFEAT - 1] * Wk[c - 1] + bk[c - 1]);
    else                 v = 0.0f;
    Kp[(size_t)row * KPAD + c] = v;
  } else {
    int cv = c - KPAD;
    float a = bv[cv];
    #pragma unroll
    for (int i = 0; i < DQIN; ++i) a += drow[i] * Wv[i * DVDIM + cv];
    Vm[(size_t)row * DVDIM + cv] = a;
  }
}

// ---------------- masked-softmax attention, fp32 WMMA, fixed m=0 ----------------
// block = 256 threads = 8 waves; wave w -> 16-query tile; grid = (T/128, N)
__global__ void __launch_bounds__(256)
attn_kernel(const float* __restrict__ Qpad,
            const float* __restrict__ Kp,
            const float* __restrict__ Vm,
            const float* __restrict__ ref_t,
            const float* __restrict__ tm,
            const float* __restrict__ log_tau,
            float* __restrict__ out) {
  __shared__ __align__(16) float sK[32 * KPAD];
  __shared__ __align__(16) float sV[32 * DVDIM];
  __shared__ __align__(16) float sP[8][16 * 16];

  const int tid  = threadIdx.x;
  const int wave = tid >> 5;
  const int lane = tid & 31;
  const int half = lane >> 4;      // 0: rows r, 1: rows r+8 (C layout)
  const int ln   = lane & 15;      // N index within C layout
  const int n    = blockIdx.y;
  const int q0   = blockIdx.x * 128 + wave * 16;

  // fold -1/tau and log2(e) into one constant: exp(-d^2/tau) = exp2(d^2 * c2)
  const float c2 = -__expf(-log_tau[0]) * 1.44269504088896340736f;

  // Q tile in A-matrix layout: lane holds M = ln; K pair = 4g + 2*half + {0,1}
  const float* Qrow = Qpad + ((size_t)n * T_DIM + q0 + ln) * KPAD;
  v2f qa[KPAD / 4];
  #pragma unroll
  for (int g = 0; g < KPAD / 4; ++g) {
    int c = 4 * g + 2 * half;
    v2f q; q[0] = Qrow[c]; q[1] = Qrow[c + 1];
    qa[g] = q;
  }

  // per-row query times (C layout rows)
  float rt[8];
  #pragma unroll
  for (int r = 0; r < 8; ++r) rt[r] = ref_t[(size_t)n * T_DIM + q0 + r + 8 * half];

  v8f acc[4];
  float lsum[8];
  #pragma unroll
  for (int t = 0; t < 4; ++t)
    #pragma unroll
    for (int r = 0; r < 8; ++r) acc[t][r] = 0.0f;
  #pragma unroll
  for (int r = 0; r < 8; ++r) lsum[r] = 0.0f;

  const float* Kbase = Kp + (size_t)n * L_DIM * KPAD;
  const float* Vbase = Vm + (size_t)n * L_DIM * DVDIM;
  const float* tmrow = tm + (size_t)n * L_DIM;
  float* pw = sP[wave];

  for (int j = 0; j < L_DIM; j += 32) {
    __syncthreads();   // protect previous iteration's LDS reads
    {
      const float4* gk = (const float4*)(Kbase + (size_t)j * KPAD);
      float4*       lk = (float4*)sK;
      #pragma unroll
      for (int i = tid; i < 32 * KPAD / 4; i += 256) lk[i] = gk[i];
      const float4* gv = (const float4*)(Vbase + (size_t)j * DVDIM);
      float4*       lv = (float4*)sV;
      #pragma unroll
      for (int i = tid; i < 32 * DVDIM / 4; i += 256) lv[i] = gv[i];
    }
    __syncthreads();

    #pragma unroll
    for (int sub = 0; sub < 2; ++sub) {
      const float* kTile = sK + sub * 16 * KPAD;
      const float* vTile = sV + sub * 16 * DVDIM;

      // S = Q * K^T  (16x16, K=68 in 17 steps of fp32 WMMA)
      v8f s = {0.f, 0.f, 0.f, 0.f, 0.f, 0.f, 0.f, 0.f};
      #pragma unroll
      for (int g = 0; g < KPAD / 4; ++g) {
        int c = 4 * g + 2 * half;
        v2f b; b[0] = kTile[ln * KPAD + c]; b[1] = kTile[ln * KPAD + c + 1];
        s = __builtin_amdgcn_wmma_f32_16x16x4_f32(false, qa[g], false, b,
                                                  (short)0, s, false, false);
      }

      const float kt = tmrow[j + sub * 16 + ln];

      // elementwise: raw v_exp_f32 (base-2), no libm underflow guard.
      // fixed zero max-shift is valid since S*mask <= 0 always.
      float e[8];
      #pragma unroll
      for (int r = 0; r < 8; ++r) {
        float d   = s[r];
        bool  msk = rt[r] >= kt;
        float ex  = __builtin_amdgcn_exp2f(d * d * c2);  // exp(-(QK)^2/tau)
        lsum[r] += msk ? ex : 1.0f;           // masked-out -> exp(0) = 1 in denom
        e[r]     = msk ? ex : 0.0f;           // masked-out -> 0 in numerator
      }

      // transpose P through per-wave LDS: C layout -> A-matrix layout
      #pragma unroll
      for (int r = 0; r < 8; ++r) pw[(r + 8 * half) * 16 + ln] = e[r];
      asm volatile("s_wait_dscnt 0" ::: "memory");
      v2f pa[4];
      #pragma unroll
      for (int g = 0; g < 4; ++g) {
        int k = 4 * g + 2 * half;
        v2f p; p[0] = pw[ln * 16 + k]; p[1] = pw[ln * 16 + k + 1];
        pa[g] = p;
      }

      // O += P * V  (4 output tiles of 16x16, K=16 each)
      #pragma unroll
      for (int t = 0; t < 4; ++t) {
        #pragma unroll
        for (int g = 0; g < 4; ++g) {
          int k = 4 * g + 2 * half;
          v2f b; b[0] = vTile[k * DVDIM + 16 * t + ln];
                 b[1] = vTile[(k + 1) * DVDIM + 16 * t + ln];
          acc[t] = __builtin_amdgcn_wmma_f32_16x16x4_f32(false, pa[g], false, b,
                                                         (short)0, acc[t], false, false);
        }
      }
    }
  }

  // one-time denominator reduction across the 16-lane row segment
  #pragma unroll
  for (int r = 0; r < 8; ++r) {
    float v = lsum[r];
    #pragma unroll
    for (int off = 1; off < 16; off <<= 1) v += __shfl_xor(v, off, 16);
    lsum[r] = v;
  }

  // normalize and store
  float* outp = out + ((size_t)n * T_DIM + q0) * DVDIM;
  #pragma unroll
  for (int r = 0; r < 8; ++r) {
    float inv  = 1.0f / lsum[r];
    int   trow = r + 8 * half;
    #pragma unroll
    for (int t = 0; t < 4; ++t)
      outp[(size_t)trow * DVDIM + 16 * t + ln] = acc[t][r] * inv;
  }
}

extern "C" void kernel_launch(void* const* d_in, const int* in_sizes, int n_in,
                              void* d_out, int out_size, void* d_ws, size_t ws_size,
                              hipStream_t stream) {
  const float* ref_data = (const float*)d_in[0];
  const float* ref_t    = (const float*)d_in[1];
  const float* mdata[3] = {(const float*)d_in[2], (const float*)d_in[4], (const float*)d_in[6]};
  const float* mt[3]    = {(const float*)d_in[3], (const float*)d_in[5], (const float*)d_in[7]};
  const float* Wq       = (const float*)d_in[8];
  const float *Wk[3], *bk[3], *Wv[3], *bv[3], *ltau[3];
  for (int m = 0; m < 3; ++m) {
    int b = 9 + 5 * m;
    Wk[m]   = (const float*)d_in[b + 0];
    bk[m]   = (const float*)d_in[b + 1];
    Wv[m]   = (const float*)d_in[b + 2];
    bv[m]   = (const float*)d_in[b + 3];
    ltau[m] = (const float*)d_in[b + 4];
  }

  float* ws = (float*)d_ws;
  const size_t qsz = (size_t)NB * T_DIM * KPAD;
  const size_t ksz = (size_t)NB * L_DIM * KPAD;
  const size_t vsz = (size_t)NB * L_DIM * DVDIM;
  float* Qpad = ws;
  float* Kp   = ws + qsz;
  float* Vm   = ws + qsz + 3 * ksz;

  {
    int tot = NB * T_DIM * KPAD;
    prep_q_kernel<<<(tot + 255) / 256, 256, 0, stream>>>(ref_data, Wq, Qpad);
  }
  for (int m = 0; m < 3; ++m) {
    int tot = NB * L_DIM * (KPAD + DVDIM);
    prep_kv_kernel<<<(tot + 255) / 256, 256, 0, stream>>>(
        mdata[m], Wk[m], bk[m], Wv[m], bv[m], Kp + m * ksz, Vm + m * vsz);
  }
  for (int m = 0; m < 3; ++m) {
    dim3 grid(T_DIM / 128, NB);
    attn_kernel<<<grid, 256, 0, stream>>>(
        Qpad, Kp + m * ksz, Vm + m * vsz, ref_t, mt[m], ltau[m],
        (float*)d_out + (size_t)m * NB * T_DIM * DVDIM);
  }
}